// GNN_75926431858906
// MI455X (gfx1250) — compile-verified
//
#include <hip/hip_runtime.h>

#define HID 96

typedef float v2f __attribute__((ext_vector_type(2)));
typedef float v8f __attribute__((ext_vector_type(8)));

// ---------------------------------------------------------------------------
// deg[dst] += 1 for every edge (self-loop handled later as +1)
// ---------------------------------------------------------------------------
__global__ void k_count_deg(const long long* __restrict__ ei, float* __restrict__ deg, int E) {
    int i = blockIdx.x * blockDim.x + threadIdx.x;
    int stride = gridDim.x * blockDim.x;
    for (int e = i; e < E; e += stride) {
        int d = (int)ei[(size_t)E + e];
        __hip_atomic_fetch_add(&deg[d], 1.0f, __ATOMIC_RELAXED, __HIP_MEMORY_SCOPE_AGENT);
    }
}

// deg -> d^{-1/2}, in place (deg >= 1 always due to self loop)
__global__ void k_dinv(float* __restrict__ deg, int n) {
    int i = blockIdx.x * blockDim.x + threadIdx.x;
    if (i < n) deg[i] = rsqrtf(deg[i] + 1.0f);
}

// ---------------------------------------------------------------------------
// H[N,96] = X[N,96] @ W[96,96] using fp32 WMMA 16x16x4.
// One wave per 16x16 output tile; W staged in LDS per block.
// A layout (16x4 f32): lanes 0-15: {K0,K1} for M=lane; lanes 16-31: {K2,K3}.
// B layout (4x16 f32): lanes 0-15: {K0,K1} for N=lane; lanes 16-31: {K2,K3}.
// C/D layout: VGPR r = row (r + 8*half), col = lane&15.
// ---------------------------------------------------------------------------
__global__ void k_gemm_wmma(const float* __restrict__ X, const float* __restrict__ W,
                            float* __restrict__ H, int nrows) {
    __shared__ float lw[HID * HID];
    for (int idx = threadIdx.x; idx < HID * HID; idx += blockDim.x) lw[idx] = W[idx];
    __syncthreads();

    int gwave = (int)((blockIdx.x * blockDim.x + threadIdx.x) >> 5);
    int lane  = threadIdx.x & 31;
    int mt = gwave / (HID / 16);
    int nt = gwave % (HID / 16);
    if (mt >= nrows / 16) return;

    int m0 = mt * 16, n0 = nt * 16;
    int half = lane >> 4;     // 0: K0/K1 slice, 1: K2/K3 slice
    int l    = lane & 15;

    const float* xrow = X + (size_t)(m0 + l) * HID;
    v8f c = {};
#pragma unroll
    for (int k0 = 0; k0 < HID; k0 += 4) {
        int ka = k0 + half * 2;
        v2f a, b;
        a.x = xrow[ka];
        a.y = xrow[ka + 1];
        b.x = lw[ka * HID + n0 + l];
        b.y = lw[(ka + 1) * HID + n0 + l];
        c = __builtin_amdgcn_wmma_f32_16x16x4_f32(false, a, false, b, (short)0, c, false, false);
    }

    float* hrow = H + (size_t)(m0 + half * 8) * HID + n0 + l;
#pragma unroll
    for (int r = 0; r < 8; ++r) hrow[(size_t)r * HID] = c[r];
}

// ---------------------------------------------------------------------------
// One wave per edge: AGG[dst] += dinv[src]*dinv[dst] * H[src]  (96 floats)
// ---------------------------------------------------------------------------
__global__ void k_scatter(const long long* __restrict__ ei, const float* __restrict__ H,
                          const float* __restrict__ dinv, float* __restrict__ AGG, int E) {
    long long gid = (long long)blockIdx.x * blockDim.x + threadIdx.x;
    long long wave = gid >> 5;
    int lane = (int)(gid & 31);
    long long nwaves = ((long long)gridDim.x * blockDim.x) >> 5;
    for (long long e = wave; e < E; e += nwaves) {
        int s = (int)ei[e];
        int d = (int)ei[(size_t)E + e];
        float norm = dinv[s] * dinv[d];
        const float* hs = H + (size_t)s * HID + lane;
        float* ad = AGG + (size_t)d * HID + lane;
        __hip_atomic_fetch_add(&ad[0],  norm * hs[0],  __ATOMIC_RELAXED, __HIP_MEMORY_SCOPE_AGENT);
        __hip_atomic_fetch_add(&ad[32], norm * hs[32], __ATOMIC_RELAXED, __HIP_MEMORY_SCOPE_AGENT);
        __hip_atomic_fetch_add(&ad[64], norm * hs[64], __ATOMIC_RELAXED, __HIP_MEMORY_SCOPE_AGENT);
    }
}

// ---------------------------------------------------------------------------
// AGG = relu(AGG + H*dinv^2 + bias)   (self-loop term + bias + relu)
// ---------------------------------------------------------------------------
__global__ void k_finalize(const float* __restrict__ H, const float* __restrict__ dinv,
                           const float* __restrict__ bias, float* __restrict__ AGG,
                           long long total) {
    long long i = (long long)blockIdx.x * blockDim.x + threadIdx.x;
    if (i >= total) return;
    int row = (int)(i / HID);
    int col = (int)(i % HID);
    float di = dinv[row];
    float v = AGG[i] + H[i] * (di * di) + bias[col];
    AGG[i] = fmaxf(v, 0.0f);
}

// ---------------------------------------------------------------------------
// Column sums -> accum[96] (atomic partials)
// ---------------------------------------------------------------------------
__global__ void k_colsum(const float* __restrict__ Hf, float* __restrict__ accum, int n) {
    int col = threadIdx.x;       // blockDim.x == 96
    float s = 0.0f;
    for (int r = blockIdx.x; r < n; r += gridDim.x) s += Hf[(size_t)r * HID + col];
    __hip_atomic_fetch_add(&accum[col], s, __ATOMIC_RELAXED, __HIP_MEMORY_SCOPE_AGENT);
}

// ---------------------------------------------------------------------------
// Head: hm = accum/N; hr = relu(hm @ lin_w + lin_b);
// out = concat(hr@q_w+q_b, hr@g_w+g_b, hr@p_w+p_b, hr@t_w+t_b)  (43 floats)
// ---------------------------------------------------------------------------
__global__ void k_heads(const float* __restrict__ accum,
                        const float* __restrict__ lin_w, const float* __restrict__ lin_b,
                        const float* __restrict__ qw, const float* __restrict__ qb,
                        const float* __restrict__ gw, const float* __restrict__ gb,
                        const float* __restrict__ pw, const float* __restrict__ pb,
                        const float* __restrict__ tw, const float* __restrict__ tb,
                        float* __restrict__ out, float invN) {
    __shared__ float hm[HID];
    __shared__ float hr[HID];
    int t = threadIdx.x;
    if (t < HID) hm[t] = accum[t] * invN;
    __syncthreads();
    if (t < HID) {
        float s = lin_b[t];
        for (int k = 0; k < HID; ++k) s += hm[k] * lin_w[k * HID + t];
        hr[t] = fmaxf(s, 0.0f);
    }
    __syncthreads();
    if (t < 43) {
        const float* w; const float* b; int cols, off, j;
        if (t < 16)      { w = qw; b = qb; cols = 16; off = 0;  j = t;      }
        else if (t < 24) { w = gw; b = gb; cols = 8;  off = 16; j = t - 16; }
        else if (t < 28) { w = pw; b = pb; cols = 4;  off = 24; j = t - 24; }
        else             { w = tw; b = tb; cols = 15; off = 28; j = t - 28; }
        float s = b[j];
        for (int k = 0; k < HID; ++k) s += hr[k] * w[k * cols + j];
        out[off + j] = s;
    }
}

// ---------------------------------------------------------------------------
extern "C" void kernel_launch(void* const* d_in, const int* in_sizes, int n_in,
                              void* d_out, int out_size, void* d_ws, size_t ws_size,
                              hipStream_t stream) {
    const float*     x     = (const float*)d_in[0];
    const long long* ei    = (const long long*)d_in[1];   // int64 [2,E]
    const float*     W1    = (const float*)d_in[2];
    const float*     b1    = (const float*)d_in[3];
    const float*     W2    = (const float*)d_in[4];
    const float*     b2    = (const float*)d_in[5];
    const float*     lin_w = (const float*)d_in[6];
    const float*     lin_b = (const float*)d_in[7];
    const float*     q_w   = (const float*)d_in[8];
    const float*     q_b   = (const float*)d_in[9];
    const float*     g_w   = (const float*)d_in[10];
    const float*     g_b   = (const float*)d_in[11];
    const float*     p_w   = (const float*)d_in[12];
    const float*     p_b   = (const float*)d_in[13];
    const float*     t_w   = (const float*)d_in[14];
    const float*     t_b   = (const float*)d_in[15];
    float* out = (float*)d_out;

    const int N = in_sizes[0] / HID;
    const int E = in_sizes[1] / 2;

    // Workspace layout
    float* dinv = (float*)d_ws;                       // N floats (deg -> dinv)
    float* A    = dinv + N;                           // N*HID
    float* B    = A + (size_t)N * HID;                // N*HID
    float* acc  = B + (size_t)N * HID;                // HID

    const long long total = (long long)N * HID;
    const int tiles   = (N / 16) * (HID / 16);
    const int gemm_bl = (tiles + 7) / 8;                               // 8 waves / 256-thr block
    const int scat_bl = (int)(((long long)E * 32 + 255) / 256);        // 1 wave / edge
    const int fin_bl  = (int)((total + 255) / 256);

    // degrees -> dinv
    hipMemsetAsync(dinv, 0, (size_t)N * sizeof(float), stream);
    k_count_deg<<<1024, 256, 0, stream>>>(ei, dinv, E);
    k_dinv<<<(N + 255) / 256, 256, 0, stream>>>(dinv, N);

    // layer 1
    k_gemm_wmma<<<gemm_bl, 256, 0, stream>>>(x, W1, A, N);
    hipMemsetAsync(B, 0, (size_t)total * sizeof(float), stream);
    k_scatter<<<scat_bl, 256, 0, stream>>>(ei, A, dinv, B, E);
    k_finalize<<<fin_bl, 256, 0, stream>>>(A, dinv, b1, B, total);     // B = relu layer1

    // layer 2 (ping-pong A/B)
    k_gemm_wmma<<<gemm_bl, 256, 0, stream>>>(B, W2, A, N);
    hipMemsetAsync(B, 0, (size_t)total * sizeof(float), stream);
    k_scatter<<<scat_bl, 256, 0, stream>>>(ei, A, dinv, B, E);
    k_finalize<<<fin_bl, 256, 0, stream>>>(A, dinv, b2, B, total);     // B = relu layer2

    // mean pool + heads
    hipMemsetAsync(acc, 0, HID * sizeof(float), stream);
    k_colsum<<<1024, HID, 0, stream>>>(B, acc, N);
    k_heads<<<1, 128, 0, stream>>>(acc, lin_w, lin_b, q_w, q_b, g_w, g_b,
                                   p_w, p_b, t_w, t_b, out, 1.0f / (float)N);
}